// DistanceEncoderSimple_23287312679168
// MI455X (gfx1250) — compile-verified
//
#include <hip/hip_runtime.h>
#include <hip/hip_bf16.h>

typedef __attribute__((ext_vector_type(2))) float v2f;
typedef __attribute__((ext_vector_type(8))) float v8f;

#define DIM 256  // embed dim, fixed by the reference

// One wave32 computes a 16-row x 256-col slab of the outer product using
// 16x V_WMMA_F32_16X16X4_F32 (rank-1 GEMM: A has dist in K=0, B has w in K=0).
// Store-bound kernel: 1.024 GB out @ 23.3 TB/s => ~44 us floor. Stores are
// non-temporal (output >> 192 MB L2) and unguarded in the fast path so each
// is a single global_store_b32 with an immediate offset.
__global__ __launch_bounds__(256) void outer_product_wmma_f32(
    const float* __restrict__ dist,
    const float* __restrict__ w,
    float* __restrict__ out,
    int n) {
  const int lane = threadIdx.x & 31;
  const int wave = blockIdx.x * (blockDim.x >> 5) + (threadIdx.x >> 5);
  const int r0   = wave << 4;                 // first of 16 rows for this wave
  if (r0 >= n) return;                        // uniform per-wave exit

  const int  ml  = lane & 15;
  const bool low = lane < 16;

  if (r0 + 16 <= n) {
    // ---------------- fast path: all 16 rows in bounds ----------------
    // A (16x4 f32): lanes 0-15 hold M=lane, VGPR0=K0; only K=0 nonzero.
    v2f a;
    a.x = low ? dist[r0 + ml] : 0.0f;
    a.y = 0.0f;

    // B tiles (4x16 f32): only K=0 row nonzero; K=0 lives in VGPR0 lanes 0-15.
    v2f b[DIM / 16];
#pragma unroll
    for (int t = 0; t < DIM / 16; ++t) {
      float wv = w[t * 16 + ml];
      b[t].x = low ? wv : 0.0f;
      b[t].y = 0.0f;
    }

    // D layout: VGPR j -> row (r0+j) lanes 0-15, row (r0+j+8) lanes 16-31;
    // column within tile = lane & 15.
    const int rowbase = r0 + (low ? 0 : 8);
    float* p = out + (size_t)rowbase * DIM + ml;  // single per-lane base addr

#pragma unroll
    for (int t = 0; t < DIM / 16; ++t) {
      v8f c = {};
      // 8 args: (neg_a, A, neg_b, B, c_mod, C, reuse_a, reuse_b)
      c = __builtin_amdgcn_wmma_f32_16x16x4_f32(
          false, a, false, b[t], (short)0, c, false, false);
#pragma unroll
      for (int j = 0; j < 8; ++j) {
        // constant byte offset: j*1024 + t*64  (fits 24-bit IOFFSET)
        __builtin_nontemporal_store(c[j], p + (size_t)j * DIM + t * 16);
      }
    }
  } else {
    // ---------------- tail path: partial last wave (plain VALU) -------
    for (int row = r0; row < n && row < r0 + 16; ++row) {
      float dv = dist[row];
      for (int c0 = lane; c0 < DIM; c0 += 32) {
        __builtin_nontemporal_store(dv * w[c0], &out[(size_t)row * DIM + c0]);
      }
    }
  }
}

extern "C" void kernel_launch(void* const* d_in, const int* in_sizes, int n_in,
                              void* d_out, int out_size, void* d_ws, size_t ws_size,
                              hipStream_t stream) {
  const float* dist = (const float*)d_in[0];   // (N,) f32
  const float* w    = (const float*)d_in[1];   // (1, 256) f32, flat 256
  float*       out  = (float*)d_out;           // (N, 256) f32

  const int n     = in_sizes[0];
  const int waves = (n + 15) / 16;             // one wave per 16 rows
  const int wpb   = 256 / 32;                  // 8 waves per 256-thread block
  const int grid  = (waves + wpb - 1) / wpb;

  outer_product_wmma_f32<<<grid, 256, 0, stream>>>(dist, w, out, n);
}